// JKNet_5995774345341
// MI455X (gfx1250) — compile-verified
//
#include <hip/hip_runtime.h>

typedef __attribute__((ext_vector_type(2))) float v2f;
typedef __attribute__((ext_vector_type(8))) float v8f;

// ---------------------------------------------------------------------------
// Utility fills (graph-capture safe zeroing of workspace regions)
// ---------------------------------------------------------------------------
__global__ void jk_fill_f32(float* __restrict__ p, long n) {
  long i = (long)blockIdx.x * blockDim.x + threadIdx.x;
  long stride = (long)gridDim.x * blockDim.x;
  for (; i < n; i += stride) p[i] = 0.0f;
}

__global__ void jk_fill_i32(int* __restrict__ p, long n) {
  long i = (long)blockIdx.x * blockDim.x + threadIdx.x;
  long stride = (long)gridDim.x * blockDim.x;
  for (; i < n; i += stride) p[i] = 0;
}

// ---------------------------------------------------------------------------
// Degree computation + symmetric norms
// ---------------------------------------------------------------------------
__global__ void jk_degree(const int* __restrict__ src, const int* __restrict__ dst,
                          int n_edges, int* __restrict__ deg_out, int* __restrict__ deg_in) {
  int i = blockIdx.x * blockDim.x + threadIdx.x;
  int stride = gridDim.x * blockDim.x;
  for (; i < n_edges; i += stride) {
    atomicAdd(&deg_out[src[i]], 1);
    atomicAdd(&deg_in[dst[i]], 1);
  }
}

__global__ void jk_norms(const int* __restrict__ deg_out, const int* __restrict__ deg_in,
                         float* __restrict__ out_norm, float* __restrict__ in_norm, int n) {
  int i = blockIdx.x * blockDim.x + threadIdx.x;
  int stride = gridDim.x * blockDim.x;
  for (; i < n; i += stride) {
    int dof = deg_out[i];
    int dif = deg_in[i];
    out_norm[i] = (dof > 0) ? rsqrtf((float)dof) : 0.0f;
    in_norm[i]  = (dif > 0) ? rsqrtf((float)dif) : 0.0f;
  }
}

// ---------------------------------------------------------------------------
// fp32 WMMA GEMM:  Y[M,Nc] = (diag(rowscale) * X[M,K]) @ Wm[K,Nc] (+ bias)
// One wave per 16x16 output tile; K-loop in steps of 4 via v_wmma_f32_16x16x4_f32.
// rowscale / bias may be null.
// ---------------------------------------------------------------------------
__global__ void jk_wmma_gemm(const float* __restrict__ X,
                             const float* __restrict__ rowscale,
                             const float* __restrict__ Wm,
                             const float* __restrict__ bias,
                             float* __restrict__ Y,
                             int M, int K, int Nc) {
  const int lane = threadIdx.x & 31;
  const int wave = threadIdx.x >> 5;
  const int wavesPerBlock = blockDim.x >> 5;
  const int ntiles = Nc >> 4;
  const int mtiles = (M + 15) >> 4;
  int tile = blockIdx.x * wavesPerBlock + wave;
  if (tile >= mtiles * ntiles) return;          // whole-wave uniform exit
  const int mt = tile / ntiles;
  const int nt = tile % ntiles;

  const int half = lane >> 4;                   // 0: K=0,1  |  1: K=2,3
  const int l16  = lane & 15;
  int row = mt * 16 + l16;                      // A-matrix row for this lane
  if (row >= M) row = M - 1;                    // clamp (no remainder for this problem)
  const int col = nt * 16 + l16;                // B/C/D column for this lane

  const float s = rowscale ? rowscale[row] : 1.0f;
  const float* __restrict__ xrow = X + (long)row * K;

  v8f acc = {};
  for (int k = 0; k < K; k += 4) {
    const int ka = k + half * 2;
    v2f a;
    a.x = xrow[ka]     * s;
    a.y = xrow[ka + 1] * s;
    v2f b;
    b.x = Wm[(long)ka       * Nc + col];
    b.y = Wm[(long)(ka + 1) * Nc + col];
    // 8 args: (neg_a, A, neg_b, B, c_mod, C, reuse_a, reuse_b)
    acc = __builtin_amdgcn_wmma_f32_16x16x4_f32(
        /*neg_a=*/false, a, /*neg_b=*/false, b,
        /*c_mod=*/(short)0, acc, /*reuse_a=*/false, /*reuse_b=*/false);
  }

  const float bv = bias ? bias[col] : 0.0f;
  // C/D layout: VGPR i -> rows (mt*16 + i) for lanes 0-15, (mt*16 + i + 8) for lanes 16-31
#pragma unroll
  for (int i = 0; i < 8; ++i) {
    const int m = mt * 16 + i + half * 8;
    if (m < M) Y[(long)m * Nc + col] = acc[i] + bv;
  }
}

// ---------------------------------------------------------------------------
// Edge scatter:  Agg[dst[e], :] += Ysrc[src[e], :]   (one block per edge,
// 128 lanes stream the feature row with coalesced loads + f32 atomics)
// ---------------------------------------------------------------------------
__global__ void jk_edge_scatter(const float* __restrict__ Ysrc,
                                const int* __restrict__ src, const int* __restrict__ dst,
                                float* __restrict__ Agg, int n_edges, int C) {
  int e = blockIdx.x;
  if (e >= n_edges) return;
  const int s = src[e];
  const int d = dst[e];
  const float* __restrict__ yrow = Ysrc + (long)s * C;
  float* __restrict__ arow = Agg + (long)d * C;
  for (int c = threadIdx.x; c < C; c += blockDim.x)
    atomicAdd(&arow[c], yrow[c]);
}

// JK pooling scatter: pooled[dst, 0:H]+=h1[src], [H:2H]+=h2[src], [2H:3H]+=h3[src]
__global__ void jk_pool_scatter(const float* __restrict__ h1, const float* __restrict__ h2,
                                const float* __restrict__ h3,
                                const int* __restrict__ src, const int* __restrict__ dst,
                                float* __restrict__ pooled, int n_edges, int H) {
  int e = blockIdx.x;
  if (e >= n_edges) return;
  const int s = src[e];
  const int d = dst[e];
  const long sb = (long)s * H;
  float* __restrict__ prow = pooled + (long)d * (3 * H);
  for (int c = threadIdx.x; c < H; c += blockDim.x) {
    atomicAdd(&prow[c],           h1[sb + c]);
    atomicAdd(&prow[H + c],       h2[sb + c]);
    atomicAdd(&prow[2 * H + c],   h3[sb + c]);
  }
}

// h = relu(agg * in_norm[row] + bias[col])
__global__ void jk_relu_norm_bias(const float* __restrict__ agg,
                                  const float* __restrict__ in_norm,
                                  const float* __restrict__ bias,
                                  float* __restrict__ h, int M, int C) {
  long total = (long)M * C;
  long i = (long)blockIdx.x * blockDim.x + threadIdx.x;
  long stride = (long)gridDim.x * blockDim.x;
  for (; i < total; i += stride) {
    const int r = (int)(i / C);
    const int c = (int)(i % C);
    float v = agg[i] * in_norm[r] + bias[c];
    h[i] = v > 0.0f ? v : 0.0f;
  }
}

// ---------------------------------------------------------------------------
// Orchestration
// ---------------------------------------------------------------------------
extern "C" void kernel_launch(void* const* d_in, const int* in_sizes, int n_in,
                              void* d_out, int out_size, void* d_ws, size_t ws_size,
                              hipStream_t stream) {
  const float* feats = (const float*)d_in[0];
  const int*   src   = (const int*)d_in[1];
  const int*   dst   = (const int*)d_in[2];
  const float* W0 = (const float*)d_in[3];
  const float* b0 = (const float*)d_in[4];
  const float* W1 = (const float*)d_in[5];
  const float* b1 = (const float*)d_in[6];
  const float* W2 = (const float*)d_in[7];
  const float* b2 = (const float*)d_in[8];
  const float* Wo = (const float*)d_in[9];
  const float* bo = (const float*)d_in[10];
  float* out = (float*)d_out;

  const int F = 128, H = 128, C = 64;
  const int n_nodes = in_sizes[0] / F;
  const int n_edges = in_sizes[1];
  (void)n_in; (void)out_size; (void)ws_size;

  // ---- workspace carve-out ----
  char* ws = (char*)d_ws;
  size_t off = 0;
  auto carve = [&](size_t bytes) {
    void* p = ws + off;
    off += (bytes + 255) & ~(size_t)255;
    return p;
  };
  int*   deg_out  = (int*)  carve((size_t)n_nodes * 4);
  int*   deg_in   = (int*)  carve((size_t)n_nodes * 4);
  float* out_norm = (float*)carve((size_t)n_nodes * 4);
  float* in_norm  = (float*)carve((size_t)n_nodes * 4);
  float* y        = (float*)carve((size_t)n_nodes * H * 4);
  float* agg      = (float*)carve((size_t)n_nodes * H * 4);
  float* h1       = (float*)carve((size_t)n_nodes * H * 4);
  float* h2       = (float*)carve((size_t)n_nodes * H * 4);
  float* h3       = (float*)carve((size_t)n_nodes * H * 4);
  float* pooled   = (float*)carve((size_t)n_nodes * 3 * H * 4);

  const int TPB = 256;
  const int FILL_BLOCKS = 4096;

  // ---- degrees + norms ----
  jk_fill_i32<<<FILL_BLOCKS, TPB, 0, stream>>>(deg_out, (long)n_nodes);
  jk_fill_i32<<<FILL_BLOCKS, TPB, 0, stream>>>(deg_in,  (long)n_nodes);
  jk_degree<<<(n_edges + TPB - 1) / TPB, TPB, 0, stream>>>(src, dst, n_edges, deg_out, deg_in);
  jk_norms<<<(n_nodes + TPB - 1) / TPB, TPB, 0, stream>>>(deg_out, deg_in, out_norm, in_norm, n_nodes);

  // ---- three GraphConv layers ----
  const int mtiles = (n_nodes + 15) / 16;
  const int wavesPerBlock = TPB / 32;

  const float* Ws[3] = {W0, W1, W2};
  const float* bs[3] = {b0, b1, b2};
  const float* xin[3] = {feats, h1, h2};
  float* hout[3] = {h1, h2, h3};

  for (int layer = 0; layer < 3; ++layer) {
    // y = (x * out_norm) @ W   (fused row scaling, fp32 WMMA)
    const int ntiles = H / 16;
    const int tiles = mtiles * ntiles;
    jk_wmma_gemm<<<(tiles + wavesPerBlock - 1) / wavesPerBlock, TPB, 0, stream>>>(
        xin[layer], out_norm, Ws[layer], nullptr, y, n_nodes, H /*K==F==H==128*/, H);

    // agg = segment_sum(y[src], dst)
    jk_fill_f32<<<FILL_BLOCKS, TPB, 0, stream>>>(agg, (long)n_nodes * H);
    jk_edge_scatter<<<n_edges, 128, 0, stream>>>(y, src, dst, agg, n_edges, H);

    // h = relu(agg * in_norm + b)
    jk_relu_norm_bias<<<FILL_BLOCKS, TPB, 0, stream>>>(agg, in_norm, bs[layer], hout[layer],
                                                       n_nodes, H);
  }

  // ---- JK concat pooling: pooled = segment_sum(concat(h1,h2,h3)[src], dst) ----
  jk_fill_f32<<<FILL_BLOCKS, TPB, 0, stream>>>(pooled, (long)n_nodes * 3 * H);
  jk_pool_scatter<<<n_edges, 128, 0, stream>>>(h1, h2, h3, src, dst, pooled, n_edges, H);

  // ---- output projection: out = pooled @ Wo + bo   [N,384] x [384,64] ----
  {
    const int ntiles = C / 16;
    const int tiles = mtiles * ntiles;
    jk_wmma_gemm<<<(tiles + wavesPerBlock - 1) / wavesPerBlock, TPB, 0, stream>>>(
        pooled, nullptr, Wo, bo, out, n_nodes, 3 * H, C);
  }
}